// Contrastive_loss_90142773609061
// MI455X (gfx1250) — compile-verified
//
#include <hip/hip_runtime.h>

typedef __attribute__((ext_vector_type(2))) float v2f;
typedef __attribute__((ext_vector_type(8))) float v8f;

#define B_N 16384
#define C_N 1000
#define D_N 128
#define CT_N 63  // ceil(1000/16) column tiles

__global__ void Closs_zero_out(float* out) { out[0] = 0.0f; }

__global__ __launch_bounds__(256) void Closs_wmma_kernel(
    const float* __restrict__ F,   // [B_N, D_N] features
    const int*   __restrict__ T,   // [B_N] targets
    const float* __restrict__ G,   // [C_N, D_N] class feature vectors
    float* __restrict__ out)       // scalar loss
{
  const int lane    = threadIdx.x & 31;
  const int wave    = threadIdx.x >> 5;
  const int lo      = lane & 15;   // N / M index within half-wave
  const int hi      = lane >> 4;   // which K-half / row-half
  const int rowBase = blockIdx.x * 16;

  // ---- Load A fragments for this 16-row feature stripe (reused across all
  // column tiles). Layout per ISA: lane(hi,lo), vgpr v -> A[lo][k0 + 2*hi + v].
  v2f afrag[32];
  const float* fRow = F + (size_t)(rowBase + lo) * D_N + 2 * hi;
  float pf2 = 0.0f;
#pragma unroll
  for (int s = 0; s < 32; ++s) {
    afrag[s] = *(const v2f*)(fRow + 4 * s);
  }
#pragma unroll
  for (int s = 0; s < 32; ++s) {
    pf2 += afrag[s].x * afrag[s].x + afrag[s].y * afrag[s].y;
  }
  // Merge the two K-halves: lane L and L^16 hold complementary halves of row lo.
  float f2full = pf2 + __shfl_xor(pf2, 16, 32);  // = ||f_{rowBase+lo}||^2

  // Epilogue needs f2 / target at row (r + 8*hi); gather once per wave.
  float f2e[8];
  int   tgt[8];
#pragma unroll
  for (int r = 0; r < 8; ++r) {
    f2e[r] = __shfl(f2full, r + 8 * hi, 32);
    tgt[r] = T[rowBase + r + 8 * hi];
  }

  float waveSum = 0.0f;

  for (int ct = wave; ct < CT_N; ct += 8) {
    const int colBase = ct * 16;
    const int j  = colBase + lo;                 // class column owned by lane
    const int jc = (j < C_N) ? j : (C_N - 1);    // clamp loads; mask later
    const float* gRow = G + (size_t)jc * D_N + 2 * hi;

    // Batch-issue all 32 B-fragment loads so they overlap in flight
    // (avoids load -> s_wait_loadcnt 0 -> wmma serialization).
    v2f bfrag[32];
#pragma unroll
    for (int s = 0; s < 32; ++s) {
      bfrag[s] = *(const v2f*)(gRow + 4 * s);    // B[2*hi+v][lo] = G[jc][k0+2*hi+v]
    }

    v8f   acc = {0.f, 0.f, 0.f, 0.f, 0.f, 0.f, 0.f, 0.f};
    float pc2 = 0.0f;
#pragma unroll
    for (int s = 0; s < 32; ++s) {
      pc2 += bfrag[s].x * bfrag[s].x + bfrag[s].y * bfrag[s].y;  // co-executes with WMMA
      // D = A x B + C  (exact fp32 path, V_WMMA_F32_16X16X4_F32)
      acc = __builtin_amdgcn_wmma_f32_16x16x4_f32(
          false, afrag[s], false, bfrag[s], (short)0, acc, false, false);
    }
    float c2full = pc2 + __shfl_xor(pc2, 16, 32);  // = ||c_j||^2

#pragma unroll
    for (int r = 0; r < 8; ++r) {
      // C/D layout: vgpr r -> row (r + 8*hi), col lo
      float dist = f2e[r] + c2full - 2.0f * acc[r];
      float h    = 1.0f - dist;
      h = h > 0.0f ? h : 0.0f;
      float m = (j < C_N && j != tgt[r]) ? 1.0f : 0.0f;  // branchless neg-mask
      waveSum += m * h;
    }
  }

  // ---- Wave32 reduction, then per-block reduction, single atomic per block.
#pragma unroll
  for (int off = 16; off > 0; off >>= 1)
    waveSum += __shfl_xor(waveSum, off, 32);

  __shared__ float smem[8];
  if (lane == 0) smem[wave] = waveSum;
  __syncthreads();
  if (threadIdx.x == 0) {
    float t = 0.0f;
#pragma unroll
    for (int q = 0; q < 8; ++q) t += smem[q];
    atomicAdd(out, t * (1.0f / (float)B_N));
  }
}

extern "C" void kernel_launch(void* const* d_in, const int* in_sizes, int n_in,
                              void* d_out, int out_size, void* d_ws, size_t ws_size,
                              hipStream_t stream) {
  (void)in_sizes; (void)n_in; (void)out_size; (void)d_ws; (void)ws_size;
  const float* features = (const float*)d_in[0];
  const int*   targets  = (const int*)d_in[1];
  const float* cvecs    = (const float*)d_in[2];
  float* out = (float*)d_out;

  Closs_zero_out<<<1, 1, 0, stream>>>(out);
  Closs_wmma_kernel<<<B_N / 16, 256, 0, stream>>>(features, targets, cvecs, out);
}